// CombinedLoss_69999376990551
// MI455X (gfx1250) — compile-verified
//
#include <hip/hip_runtime.h>
#include <math.h>

typedef __attribute__((ext_vector_type(2))) float v2f;
typedef __attribute__((ext_vector_type(8))) float v8f;

#define NB     2
#define NPTS   8192
#define NEDGE  24576
#define FEAT   6
#define EPSF   1e-12f
#define CHUNK  4096          // Y rows staged in LDS per phase (64 KB of float4)

// ---- workspace layout (in floats) ----
// [0]        total loss accumulator
// [8..13]    normsq_edge  [b*3+d]
// [16..21]   normsq_normal[b*3+d]
// [32 ..)    argmin (int), NB*NPTS entries
// [32+NB*NPTS ..) 4 squared-norm arrays, each NB*NPTS
#define WS_ACC 0
#define WS_NSE 8
#define WS_NSN 16
#define WS_ARG 32
#define WS_SQ  (32 + NB * NPTS)

__global__ void k_init(const float* __restrict__ gts, const float* __restrict__ preds,
                       float* __restrict__ ws) {
    int i = blockIdx.x * blockDim.x + threadIdx.x;
    if (i < 32) ws[i] = 0.0f;
    const int total = NB * NPTS;
    if (i < total) {
        const float* g = gts + (size_t)i * FEAT;
        const float* p = preds + (size_t)i * FEAT;
        float* sq = ws + WS_SQ;
        sq[0 * total + i] = g[0] * g[0] + g[1] * g[1] + g[2] * g[2];
        sq[1 * total + i] = p[0] * p[0] + p[1] * p[1] + p[2] * p[2];
        sq[2 * total + i] = g[3] * g[3] + g[4] * g[4] + g[5] * g[5];
        sq[3 * total + i] = p[3] * p[3] + p[4] * p[4] + p[5] * p[5];
    }
}

// Row-min (and optional argmin) of pairwise sqdist between X rows and Y rows
// over 3 features at column `off`. One wave owns 16 rows. Y is staged through
// LDS in 4096-row chunks as float4 [x,y,z,sqnorm]; the cross term is one
// V_WMMA_F32_16X16X4_F32 per 16x16 tile.
template <bool TRACK>
__global__ void k_nnmin(const float* __restrict__ X, const float* __restrict__ Y,
                        const float* __restrict__ sqX, const float* __restrict__ sqY,
                        int off, int* __restrict__ argmin_out,
                        float* __restrict__ acc, float weight) {
    __shared__ float4 lds[CHUNK];

    const int b     = blockIdx.y;
    const int lane  = threadIdx.x & 31;
    const int wave  = threadIdx.x >> 5;
    const int n0    = blockIdx.x * 128 + wave * 16;   // 8 waves * 16 rows
    const int l15   = lane & 15;
    const int halfU = lane >> 4;

    const float* Xb   = X + (size_t)b * NPTS * FEAT;
    const float* Yb   = Y + (size_t)b * NPTS * FEAT;
    const float* sqXb = sqX + b * NPTS;
    const float* sqYb = sqY + b * NPTS;

    // A tile 16x4 (K=3 padded): lanes 0-15 hold K=0,1; lanes 16-31 hold K=2,(3=0)
    v2f a;
    {
        const float* xr = Xb + (size_t)(n0 + l15) * FEAT + off;
        float x0 = xr[0], x1 = xr[1], x2 = xr[2];        // branchless
        a.x = (halfU == 0) ? x0 : x2;
        a.y = (halfU == 0) ? x1 : 0.0f;
    }

    // squared norms of my 8 rows (C/D layout: VGPR v -> row v + 8*half)
    float ra[8];
#pragma unroll
    for (int v = 0; v < 8; ++v) ra[v] = sqXb[n0 + v + 8 * halfU];

    // track min of (ry - 2*dot); add ra at the end (row-constant)
    float bestVal[8];
    int   bestIdx[8];
#pragma unroll
    for (int v = 0; v < 8; ++v) { bestVal[v] = 3.0e38f; bestIdx[v] = 0; }

    for (int chunk = 0; chunk < NPTS; chunk += CHUNK) {
        __syncthreads();
        // cooperative stage of Y chunk into LDS
        for (int i = threadIdx.x; i < CHUNK; i += 256) {
            const float* yr = Yb + (size_t)(chunk + i) * FEAT + off;
            float4 vv;
            vv.x = yr[0]; vv.y = yr[1]; vv.z = yr[2];
            vv.w = sqYb[chunk + i];
            lds[i] = vv;
        }
        __syncthreads();

#pragma unroll 2
        for (int t = 0; t < CHUNK; t += 16) {
            float4 yd = lds[t + l15];                    // ds_load_b128, bank-free
            v2f bt;
            bt.x = (halfU == 0) ? yd.x : yd.z;
            bt.y = (halfU == 0) ? yd.y : 0.0f;
            float ry = yd.w;

            v8f c = {};
            c = __builtin_amdgcn_wmma_f32_16x16x4_f32(false, a, false, bt,
                                                      (short)0, c, false, false);
            const int mIdx = chunk + t + l15;
#pragma unroll
            for (int v = 0; v < 8; ++v) {
                float d = fmaf(-2.0f, c[v], ry);
                if (TRACK) {
                    if (d < bestVal[v]) { bestVal[v] = d; bestIdx[v] = mIdx; }
                } else {
                    bestVal[v] = fminf(bestVal[v], d);
                }
            }
        }
    }

    // butterfly min across the 16 lanes sharing each row
#pragma unroll
    for (int mask = 8; mask >= 1; mask >>= 1) {
#pragma unroll
        for (int v = 0; v < 8; ++v) {
            float ov = __shfl_xor(bestVal[v], mask, 32);
            if (TRACK) {
                int oi = __shfl_xor(bestIdx[v], mask, 32);
                if (ov < bestVal[v]) { bestVal[v] = ov; bestIdx[v] = oi; }
            } else {
                bestVal[v] = fminf(bestVal[v], ov);
            }
        }
    }

    if (TRACK && l15 == 0) {
#pragma unroll
        for (int v = 0; v < 8; ++v)
            argmin_out[b * NPTS + n0 + v + 8 * halfU] = bestIdx[v];
    }

    float s = 0.0f;
#pragma unroll
    for (int v = 0; v < 8; ++v) s += ra[v] + bestVal[v];
    s += __shfl_xor(s, 16, 32);          // combine both 8-row halves
    if (lane == 0) atomicAdd(acc, weight * s);
}

// Pass 1 over edges: edge L1 loss + per-batch per-axis sums of squares for the
// (axis=1!) normalization in _unit_axis1.
__global__ void k_edge1(const float* __restrict__ preds,
                        const float* __restrict__ gts_normals,
                        const int* __restrict__ edges,
                        const int* __restrict__ argmin,
                        float* __restrict__ ws) {
    __shared__ float sh[7];
    const int t = threadIdx.x;
    if (t < 7) sh[t] = 0.0f;
    __syncthreads();

    const int i = blockIdx.x * blockDim.x + t;   // [0, NB*NEDGE); NEDGE % 256 == 0
    const int b = i / NEDGE;
    const int e = i % NEDGE;
    const int e0 = edges[2 * e + 0];
    const int e1 = edges[2 * e + 1];
    const float* pb = preds + (size_t)b * NPTS * FEAT;
    const float* p0 = pb + (size_t)e0 * FEAT;
    const float* p1 = pb + (size_t)e1 * FEAT;
    float d0 = p0[0] - p1[0], d1 = p0[1] - p1[1], d2 = p0[2] - p1[2];
    float el = fabsf(d0) + fabsf(d1) + fabsf(d2);
    float t0 = truncf(d0), t1 = truncf(d1), t2 = truncf(d2);
    const int idx = argmin[b * NPTS + e0];
    const float* nv = gts_normals + ((size_t)b * NPTS + idx) * 3;
    float n0v = nv[0], n1v = nv[1], n2v = nv[2];

    atomicAdd(&sh[0], el);
    atomicAdd(&sh[1], t0 * t0);  atomicAdd(&sh[2], t1 * t1);  atomicAdd(&sh[3], t2 * t2);
    atomicAdd(&sh[4], n0v * n0v); atomicAdd(&sh[5], n1v * n1v); atomicAdd(&sh[6], n2v * n2v);
    __syncthreads();

    if (t == 0) atomicAdd(ws + WS_ACC, (300.0f / (float)NEDGE) * sh[0]);
    if (t >= 1 && t < 4) atomicAdd(ws + WS_NSE + b * 3 + (t - 1), sh[t]);
    if (t >= 4 && t < 7) atomicAdd(ws + WS_NSN + b * 3 + (t - 4), sh[t]);
}

// Pass 2: cosine term with the axis-1 normalizers.
__global__ void k_edge2(const float* __restrict__ preds,
                        const float* __restrict__ gts_normals,
                        const int* __restrict__ edges,
                        const int* __restrict__ argmin,
                        float* __restrict__ ws) {
    __shared__ float sh;
    const int t = threadIdx.x;
    if (t == 0) sh = 0.0f;
    __syncthreads();

    const int i = blockIdx.x * blockDim.x + t;
    const int b = i / NEDGE;
    const int e = i % NEDGE;
    const int e0 = edges[2 * e + 0];
    const int e1 = edges[2 * e + 1];
    const float* pb = preds + (size_t)b * NPTS * FEAT;
    const float* p0 = pb + (size_t)e0 * FEAT;
    const float* p1 = pb + (size_t)e1 * FEAT;
    float t0 = truncf(p0[0] - p1[0]);
    float t1 = truncf(p0[1] - p1[1]);
    float t2 = truncf(p0[2] - p1[2]);
    const int idx = argmin[b * NPTS + e0];
    const float* nv = gts_normals + ((size_t)b * NPTS + idx) * 3;

    float ie0 = 1.0f / fmaxf(sqrtf(ws[WS_NSE + b * 3 + 0]), EPSF);
    float ie1 = 1.0f / fmaxf(sqrtf(ws[WS_NSE + b * 3 + 1]), EPSF);
    float ie2 = 1.0f / fmaxf(sqrtf(ws[WS_NSE + b * 3 + 2]), EPSF);
    float in0 = 1.0f / fmaxf(sqrtf(ws[WS_NSN + b * 3 + 0]), EPSF);
    float in1 = 1.0f / fmaxf(sqrtf(ws[WS_NSN + b * 3 + 1]), EPSF);
    float in2 = 1.0f / fmaxf(sqrtf(ws[WS_NSN + b * 3 + 2]), EPSF);

    float cosv = fabsf(nv[0] * in0 * t0 * ie0 +
                       nv[1] * in1 * t1 * ie1 +
                       nv[2] * in2 * t2 * ie2);
    atomicAdd(&sh, cosv);
    __syncthreads();
    if (t == 0) atomicAdd(ws + WS_ACC, (0.5f / (float)NEDGE) * sh);
}

__global__ void k_final(const float* __restrict__ ws, float* __restrict__ out) {
    out[0] = ws[WS_ACC];
}

extern "C" void kernel_launch(void* const* d_in, const int* in_sizes, int n_in,
                              void* d_out, int out_size, void* d_ws, size_t ws_size,
                              hipStream_t stream) {
    const float* gts         = (const float*)d_in[0];
    const float* preds       = (const float*)d_in[1];
    const float* gts_normals = (const float*)d_in[2];
    const int*   edges       = (const int*)d_in[3];

    float* ws     = (float*)d_ws;
    int*   argmin = (int*)(ws + WS_ARG);
    float* sq     = ws + WS_SQ;
    float* acc    = ws + WS_ACC;
    const int total = NB * NPTS;

    k_init<<<(total + 255) / 256, 256, 0, stream>>>(gts, preds, ws);

    dim3 grid(NPTS / 128, NB);
    // points, gts rows -> dist_s2f (mean over N, *0.55*3000) + argmin for normals
    k_nnmin<true><<<grid, 256, 0, stream>>>(gts, preds, sq + 0 * total, sq + 1 * total,
                                            0, argmin, acc, 3000.0f * 0.55f / (float)NPTS);
    // points, preds rows -> dist_f2s (mean over N, *3000)
    k_nnmin<false><<<grid, 256, 0, stream>>>(preds, gts, sq + 1 * total, sq + 0 * total,
                                             0, nullptr, acc, 3000.0f / (float)NPTS);
    // colors, min over axis 2 (per gts row)
    k_nnmin<false><<<grid, 256, 0, stream>>>(gts, preds, sq + 2 * total, sq + 3 * total,
                                             3, nullptr, acc, 1.0f);
    // colors, min over axis 1 (per preds row)
    k_nnmin<false><<<grid, 256, 0, stream>>>(preds, gts, sq + 3 * total, sq + 2 * total,
                                             3, nullptr, acc, 1.0f);

    const int eth = NB * NEDGE;
    k_edge1<<<eth / 256, 256, 0, stream>>>(preds, gts_normals, edges, argmin, ws);
    k_edge2<<<eth / 256, 256, 0, stream>>>(preds, gts_normals, edges, argmin, ws);
    k_final<<<1, 1, 0, stream>>>(ws, (float*)d_out);
}